// MF_10943576670363
// MI455X (gfx1250) — compile-verified
//
#include <hip/hip_runtime.h>

// Median blur 3x3, zero padding, [16,3,512,512] fp32 -> same shape.
// Bandwidth-bound (~100 MB total traffic => ~4.3us floor @ 23.3 TB/s).
// Uses CDNA5 async global->LDS copies (ASYNCcnt) to stage halo tiles.

#define WIDTH   512
#define HEIGHT  512
#define PLANE   (WIDTH * HEIGHT)   // 262144
#define TILE_W  64
#define TILE_H  16
#define HALO_W  (TILE_W + 2)       // 66
#define HALO_H  (TILE_H + 2)       // 18
#define PITCH   68                 // LDS row pitch in floats
#define NTHREADS 256

__device__ __forceinline__ void mm(float &a, float &b) {
    float t = fminf(a, b);
    b = fmaxf(a, b);
    a = t;
}

// Classic 19-compare-exchange median-of-9 network; result in p4.
__device__ __forceinline__ float median9(float p0, float p1, float p2,
                                         float p3, float p4, float p5,
                                         float p6, float p7, float p8) {
    mm(p1, p2); mm(p4, p5); mm(p7, p8);
    mm(p0, p1); mm(p3, p4); mm(p6, p7);
    mm(p1, p2); mm(p4, p5); mm(p7, p8);
    mm(p0, p3); mm(p5, p8); mm(p4, p7);
    mm(p3, p6); mm(p1, p4); mm(p2, p5);
    mm(p4, p7); mm(p4, p2); mm(p6, p4);
    mm(p4, p2);
    return p4;
}

__global__ __launch_bounds__(NTHREADS) void median3x3_async_kernel(
        const float* __restrict__ in, float* __restrict__ out) {
    __shared__ float tile[PITCH * HALO_H];

    const int tid   = threadIdx.x;
    const int x0    = blockIdx.x * TILE_W;
    const int y0    = blockIdx.y * TILE_H;
    const int plane = blockIdx.z;

    const float* pin = in + (size_t)plane * PLANE;

    // LDS byte address of tile[0]: low 32 bits of the generic pointer are the
    // LDS offset (flat->LDS aperture mapping truncates to addr[31:0]).
    const unsigned lds_base = (unsigned)(unsigned long long)&tile[0];

    // ---- Stage 66x18 halo into LDS ----------------------------------------
    // In-bounds elements: async DMA global->LDS (no VGPR round trip).
    // Out-of-bounds elements: zero pad via normal LDS store.
    #pragma unroll
    for (int i = tid; i < HALO_W * HALO_H; i += NTHREADS) {
        const int hy = i / HALO_W;
        const int hx = i - hy * HALO_W;
        const int gx = x0 + hx - 1;
        const int gy = y0 + hy - 1;
        const unsigned lds_off = lds_base + (unsigned)(hy * PITCH + hx) * 4u;
        if ((unsigned)gx < (unsigned)WIDTH && (unsigned)gy < (unsigned)HEIGHT) {
            const unsigned goff = (unsigned)(gy * WIDTH + gx) * 4u;
            // GVS form: mem_addr = SGPR64 base + per-lane u32 byte offset.
            asm volatile("global_load_async_to_lds_b32 %0, %1, %2"
                         :: "v"(lds_off), "v"(goff), "s"(pin)
                         : "memory");
        } else {
            tile[hy * PITCH + hx] = 0.0f;
        }
    }

    // Wait for this wave's async copies, then make LDS visible block-wide.
    asm volatile("s_wait_asynccnt 0" ::: "memory");
    __syncthreads();

    // ---- Compute: each thread does 4 vertically consecutive pixels --------
    const int tx = tid & (TILE_W - 1);   // 0..63
    const int ty = tid >> 6;             // 0..3
    const int yy = ty * 4;               // first output row within tile

    float* pout = out + (size_t)plane * PLANE;

    int rb = yy * PITCH + tx;            // halo row yy, halo col tx (= window col 0)
    float a0 = tile[rb], a1 = tile[rb + 1], a2 = tile[rb + 2];
    rb += PITCH;
    float b0 = tile[rb], b1 = tile[rb + 1], b2 = tile[rb + 2];

    #pragma unroll
    for (int r = 0; r < 4; ++r) {
        rb += PITCH;
        const float c0 = tile[rb], c1 = tile[rb + 1], c2 = tile[rb + 2];
        const float m = median9(a0, a1, a2, b0, b1, b2, c0, c1, c2);
        pout[(size_t)(y0 + yy + r) * WIDTH + (x0 + tx)] = m;
        a0 = b0; a1 = b1; a2 = b2;
        b0 = c0; b1 = c1; b2 = c2;
    }
}

extern "C" void kernel_launch(void* const* d_in, const int* in_sizes, int n_in,
                              void* d_out, int out_size, void* d_ws, size_t ws_size,
                              hipStream_t stream) {
    (void)n_in; (void)d_ws; (void)ws_size; (void)out_size;
    const float* image = (const float*)d_in[0];   // d_in[1] (cover_image) unused
    float* out = (float*)d_out;

    const int planes = in_sizes[0] / PLANE;       // B*C = 48
    dim3 grid(WIDTH / TILE_W, HEIGHT / TILE_H, planes);
    dim3 block(NTHREADS);
    median3x3_async_kernel<<<grid, block, 0, stream>>>(image, out);
}